// GeoEncoder_27547920237039
// MI455X (gfx1250) — compile-verified
//
#include <hip/hip_runtime.h>

typedef __attribute__((ext_vector_type(2))) float v2f;
typedef __attribute__((ext_vector_type(8))) float v8f;

constexpr int RES     = 256;
constexpr int RANK    = 12;
constexpr int FDIM    = 36;   // 3 * RANK
constexpr int ODIM    = 8;
constexpr int FSTRIDE = 44;   // feats LDS row stride (floats): 176B rows, 16B aligned, conflict-free
constexpr int OSTRIDE = 12;   // outT LDS row stride (floats): 48B rows, 16B aligned
constexpr int BLOCK   = 256;
constexpr int WAVES   = BLOCK / 32;

__device__ __forceinline__ float lerp2(float v00, float v01, float v10, float v11,
                                       float wx, float wy) {
    float top = fmaf(wx, v01 - v00, v00);
    float bot = fmaf(wx, v11 - v10, v10);
    return fmaf(wy, bot - top, top);
}

// Bilinear sample (align_corners=False, border padding) of a [256,256,12] plane.
// Writes 12 floats to f (LDS), 16B-aligned.
__device__ __forceinline__ void sample_plane12(const float* __restrict__ plane,
                                               float x, float y,
                                               float* __restrict__ f) {
    // ((x+1)*W - 1)/2 = 128*x + 127.5, then border clamp
    float ix = fminf(fmaxf(fmaf(x, 128.0f, 127.5f), 0.0f), 255.0f);
    float iy = fminf(fmaxf(fmaf(y, 128.0f, 127.5f), 0.0f), 255.0f);
    float x0f = floorf(ix), y0f = floorf(iy);
    float wx = ix - x0f, wy = iy - y0f;
    int x0 = (int)x0f, y0 = (int)y0f;
    int x1 = x0 + 1; x1 = (x1 > RES - 1) ? (RES - 1) : x1;
    int y1 = y0 + 1; y1 = (y1 > RES - 1) ? (RES - 1) : y1;
    const float4* p00 = reinterpret_cast<const float4*>(plane + (y0 * RES + x0) * RANK);
    const float4* p01 = reinterpret_cast<const float4*>(plane + (y0 * RES + x1) * RANK);
    const float4* p10 = reinterpret_cast<const float4*>(plane + (y1 * RES + x0) * RANK);
    const float4* p11 = reinterpret_cast<const float4*>(plane + (y1 * RES + x1) * RANK);
#pragma unroll
    for (int j = 0; j < 3; ++j) {
        float4 a = p00[j], b = p01[j], c = p10[j], d = p11[j];
        float4 r;
        r.x = lerp2(a.x, b.x, c.x, d.x, wx, wy);
        r.y = lerp2(a.y, b.y, c.y, d.y, wx, wy);
        r.z = lerp2(a.z, b.z, c.z, d.z, wx, wy);
        r.w = lerp2(a.w, b.w, c.w, d.w, wx, wy);
        *reinterpret_cast<float4*>(f + 4 * j) = r;
    }
}

__global__ __launch_bounds__(BLOCK) void triplane_wmma_kernel(
        const float* __restrict__ coords,
        const float* __restrict__ pxy,
        const float* __restrict__ pxz,
        const float* __restrict__ pyz,
        const float* __restrict__ projw,   // [8, 36]
        const float* __restrict__ projb,   // [8]
        float* __restrict__ out,           // [N, 8]
        int N) {
    __shared__ float feats[WAVES][32][FSTRIDE];
    __shared__ float wlds[16][FDIM];               // proj_w zero-padded to 16 rows
    __shared__ float blds[16];                     // bias zero-padded to 16
    __shared__ float outT[WAVES][16][OSTRIDE];     // per-wave store-transpose tile

    const int tid  = threadIdx.x;
    const int lane = tid & 31;
    const int wid  = tid >> 5;

    // ---- Stage projection weights + bias into LDS (coalesced, once per block) ----
    float* wflat = &wlds[0][0];
    for (int i = tid; i < 16 * FDIM; i += BLOCK)
        wflat[i] = (i < ODIM * FDIM) ? projw[i] : 0.0f;
    if (tid < 16) blds[tid] = (tid < ODIM) ? projb[tid] : 0.0f;

    // ---- Phase 1: per-lane bilinear gathers -> 36 features in LDS ----
    int gid = blockIdx.x * BLOCK + tid;
    int sid = (gid < N) ? gid : (N - 1);   // clamp: EXEC must stay all-1s for WMMA

    float cx = coords[sid * 3 + 0];
    float cy = coords[sid * 3 + 1];
    float cz = coords[sid * 3 + 2];
    cx = fminf(fmaxf(cx, -1.0f), 1.0f);
    cy = fminf(fmaxf(cy, -1.0f), 1.0f);
    cz = fminf(fmaxf(cz, -1.0f), 1.0f);

    float* f = &feats[wid][lane][0];
    sample_plane12(pxy, cx, cy, f);
    sample_plane12(pxz, cx, cz, f + RANK);
    sample_plane12(pyz, cy, cz, f + 2 * RANK);

    __syncthreads();

    // ---- Phase 2: projection via V_WMMA_F32_16X16X4_F32 ----
    // D[m,n] = sum_k A[m,k]*B[k,n] + C[m,n];  C preloaded with bias, B columns 8..15 are zero.
    const int m  = lane & 15;     // A row index == D column index for this lane
    const int kh = lane >> 4;     // K half: lanes 0-15 -> K=0,1 ; lanes 16-31 -> K=2,3

    v2f bmat[9];
#pragma unroll
    for (int kk = 0; kk < 9; ++kk)
        bmat[kk] = *reinterpret_cast<const v2f*>(&wlds[m][4 * kk + 2 * kh]);
    const float bn = blds[m];

    const int  blockBase = blockIdx.x * BLOCK + wid * 32;
    const bool fullwave  = (blockBase + 32) <= N;   // wave-uniform

#pragma unroll
    for (int t = 0; t < 2; ++t) {     // two 16-point tiles per wave
        v8f acc;
#pragma unroll
        for (int v = 0; v < 8; ++v) acc[v] = bn;    // bias folded into C
#pragma unroll
        for (int kk = 0; kk < 9; ++kk) {
            v2f a = *reinterpret_cast<const v2f*>(&feats[wid][t * 16 + m][4 * kk + 2 * kh]);
            acc = __builtin_amdgcn_wmma_f32_16x16x4_f32(
                /*neg_a=*/false, a, /*neg_b=*/false, bmat[kk],
                /*c_mod=*/(short)0, acc, /*reuse_a=*/false, /*reuse_b=*/false);
        }

        if (fullwave) {
            // Fast path: transpose D through LDS, emit one coalesced 512B store per tile.
            if (m < ODIM) {
#pragma unroll
                for (int v = 0; v < 8; ++v)
                    outT[wid][v + 8 * kh][m] = fminf(fmaxf(acc[v], -10.0f), 10.0f);
            }
            __builtin_amdgcn_wave_barrier();   // same-wave DS ordering (no codegen)
            const int r  = lane >> 1;          // point row within tile
            const int cc = (lane & 1) * 4;     // float4 chunk within row
            float4 o = *reinterpret_cast<const float4*>(&outT[wid][r][cc]);
            *reinterpret_cast<float4*>(&out[(blockBase + t * 16 + r) * ODIM + cc]) = o;
            __builtin_amdgcn_wave_barrier();   // keep t=1 LDS writes after t=0 reads
        } else {
            // Tail path (only the final partial wave of the grid).
            if (m < ODIM) {
#pragma unroll
                for (int v = 0; v < 8; ++v) {
                    const int pt = blockBase + t * 16 + v + 8 * kh;
                    if (pt < N)
                        out[pt * ODIM + m] = fminf(fmaxf(acc[v], -10.0f), 10.0f);
                }
            }
        }
    }
}

extern "C" void kernel_launch(void* const* d_in, const int* in_sizes, int n_in,
                              void* d_out, int out_size, void* d_ws, size_t ws_size,
                              hipStream_t stream) {
    const float* coords = (const float*)d_in[0];
    const float* pxy    = (const float*)d_in[1];
    const float* pxz    = (const float*)d_in[2];
    const float* pyz    = (const float*)d_in[3];
    const float* projw  = (const float*)d_in[4];
    const float* projb  = (const float*)d_in[5];
    float* out = (float*)d_out;

    const int N = in_sizes[0] / 3;
    const int blocks = (N + BLOCK - 1) / BLOCK;
    triplane_wmma_kernel<<<blocks, BLOCK, 0, stream>>>(
        coords, pxy, pxz, pyz, projw, projb, out, N);
}